// SpamClassifierLstmPosUniversal_87479893884952
// MI455X (gfx1250) — compile-verified
//
#include <hip/hip_runtime.h>
#include <math.h>

// Problem constants (from reference)
#define E_DIM  300
#define H_DIM  512
#define G4     2048      // 4*H
#define T_LEN  300
#define NEXP   13

typedef float v2f __attribute__((ext_vector_type(2)));
typedef float v8f __attribute__((ext_vector_type(8)));

// Workspace layout (byte offsets into d_ws)
#define WS_COUNTS 0        // NEXP ints
#define WS_LISTS  1024     // NEXP*T ints   (15,600 B)
#define WS_XEMB   20480    // T*E floats    (360,000 B)
#define WS_XGATES 393216   // T*4H floats   (2,457,600 B)

__device__ __forceinline__ float sigmoidf_(float x) {
    return 1.0f / (1.0f + __expf(-x));
}

// -------- Phase 0: bucket timesteps by expert, gather embeddings --------
__global__ __launch_bounds__(256) void setup_kernel(const int* __restrict__ x,
                                                    const int* __restrict__ tags,
                                                    const float* __restrict__ emb,
                                                    int* __restrict__ counts,
                                                    int* __restrict__ lists,
                                                    float* __restrict__ xemb)
{
    const int tid = threadIdx.x;
    if (tid < NEXP) counts[tid] = 0;
    __syncthreads();
    for (int t = tid; t < T_LEN; t += blockDim.x) {
        int e = tags[t];
        int slot = atomicAdd(&counts[e], 1);
        lists[e * T_LEN + slot] = t;
    }
    for (int idx = tid; idx < T_LEN * E_DIM; idx += blockDim.x) {
        int t = idx / E_DIM;
        int k = idx - t * E_DIM;
        xemb[idx] = emb[(long)x[t] * E_DIM + k];
    }
}

// -------- Phase 1: grouped GEMM  xgates[t] = W_ih[tag[t]] @ x_t + b_ih + b_hh
// One wave per (expert e, 16-row tile of the 2048 gate rows, 16-col tile of
// that expert's timestep list). Uses V_WMMA_F32_16X16X4_F32 (exact f32 path).
__global__ __launch_bounds__(32) void xproj_kernel(const float* __restrict__ W_ih,
                                                   const float* __restrict__ b_ih,
                                                   const float* __restrict__ b_hh,
                                                   const int* __restrict__ counts,
                                                   const int* __restrict__ lists,
                                                   const float* __restrict__ xemb,
                                                   float* __restrict__ xgates)
{
    const int mtile = blockIdx.x;   // 0..127 (2048/16)
    const int ntile = blockIdx.y;   // 0..18  (ceil(300/16))
    const int e     = blockIdx.z;   // 0..12
    const int cnt   = counts[e];
    if (ntile * 16 >= cnt) return;  // wave-uniform

    const int lane = threadIdx.x;   // 0..31
    const int half = lane >> 4;     // 0 or 1
    const int l15  = lane & 15;

    // A (16x4 f32, ISA layout): lane -> M = lane%16; v0/v1 hold K = 2*half, 2*half+1
    const int jrow = mtile * 16 + l15;
    const float* Wrow = W_ih + ((long)e * G4 + jrow) * E_DIM + 2 * half;

    // B (4x16 f32, ISA layout): lane -> N = lane%16; v0/v1 hold K = 2*half, 2*half+1
    const int nslot = ntile * 16 + l15;
    const int tcol  = lists[e * T_LEN + (nslot < cnt ? nslot : ntile * 16)];
    const float* Xcol = xemb + (long)tcol * E_DIM + 2 * half;

    v8f acc = {};
    for (int k = 0; k < E_DIM; k += 4) {      // 75 iterations, K=300
        v2f a = *(const v2f*)(Wrow + k);
        v2f b = *(const v2f*)(Xcol + k);
        acc = __builtin_amdgcn_wmma_f32_16x16x4_f32(
            /*neg_a=*/false, a, /*neg_b=*/false, b,
            /*c_mod=*/(short)0, acc, /*reuse_a=*/false, /*reuse_b=*/false);
    }

    // D (16x16 f32 C/D layout): VGPR i, lanes0-15 -> M=i, lanes16-31 -> M=i+8; N=lane%16
    if (nslot < cnt) {
        #pragma unroll
        for (int i = 0; i < 8; ++i) {
            int j = mtile * 16 + i + 8 * half;
            xgates[(long)tcol * G4 + j] = acc[i] + b_ih[e * G4 + j] + b_hh[e * G4 + j];
        }
    }
}

// -------- Phase 2: sequential LSTM recurrence (one WGP, state in LDS) --------
__global__ __launch_bounds__(1024) void lstm_seq_kernel(const int* __restrict__ tags,
                                                        const float* __restrict__ h0,
                                                        const float* __restrict__ c0,
                                                        const float* __restrict__ W_hh,
                                                        const float* __restrict__ xgates,
                                                        const float* __restrict__ fc_w,
                                                        const float* __restrict__ fc_b,
                                                        float* __restrict__ out)
{
    __shared__ float h_s[H_DIM];
    __shared__ float c_s[H_DIM];
    __shared__ float g_s[G4];
    const int tid = threadIdx.x;

    if (tid < H_DIM) { h_s[tid] = h0[tid]; c_s[tid] = c0[tid]; }
    __syncthreads();

    for (int t = 0; t < T_LEN; ++t) {
        const int e = tags[t];
        const float* Wh = W_hh + (long)e * G4 * H_DIM;
        const float* xg = xgates + (long)t * G4;

        // gates[j] = xgates[t][j] + sum_k W_hh[e][j][k] * h[k]
        for (int j = tid; j < G4; j += 1024) {
            const float4* w4 = (const float4*)(Wh + (long)j * H_DIM);
            float acc = xg[j];
            #pragma unroll 8
            for (int k4 = 0; k4 < H_DIM / 4; ++k4) {
                float4 w = w4[k4];
                const float* hh = &h_s[k4 * 4];
                acc += w.x * hh[0] + w.y * hh[1] + w.z * hh[2] + w.w * hh[3];
            }
            g_s[j] = acc;
        }
        __syncthreads();

        if (tid < H_DIM) {
            float ig = sigmoidf_(g_s[tid]);
            float fg = sigmoidf_(g_s[H_DIM + tid]);
            float gg = tanhf(g_s[2 * H_DIM + tid]);
            float og = sigmoidf_(g_s[3 * H_DIM + tid]);
            float c  = fg * c_s[tid] + ig * gg;
            c_s[tid] = c;
            h_s[tid] = og * tanhf(c);
        }
        __syncthreads();
    }

    // out = sigmoid(h @ fc_w.T + fc_b)  (O == 1)
    if (tid < H_DIM) g_s[tid] = h_s[tid] * fc_w[tid];
    __syncthreads();
    for (int s = H_DIM / 2; s > 0; s >>= 1) {
        if (tid < s) g_s[tid] += g_s[tid + s];
        __syncthreads();
    }
    if (tid == 0) out[0] = sigmoidf_(g_s[0] + fc_b[0]);
    if (tid < H_DIM) {
        out[1 + tid] = h_s[tid];             // h
        out[1 + H_DIM + tid] = c_s[tid];     // c
    }
}

extern "C" void kernel_launch(void* const* d_in, const int* in_sizes, int n_in,
                              void* d_out, int out_size, void* d_ws, size_t ws_size,
                              hipStream_t stream) {
    const int*   x    = (const int*)d_in[0];
    const int*   tags = (const int*)d_in[1];
    const float* h0   = (const float*)d_in[2];
    const float* c0   = (const float*)d_in[3];
    const float* emb  = (const float*)d_in[4];
    const float* W_ih = (const float*)d_in[5];
    const float* W_hh = (const float*)d_in[6];
    const float* b_ih = (const float*)d_in[7];
    const float* b_hh = (const float*)d_in[8];
    const float* fc_w = (const float*)d_in[9];
    const float* fc_b = (const float*)d_in[10];
    float* out = (float*)d_out;

    char* ws = (char*)d_ws;
    int*   counts = (int*)(ws + WS_COUNTS);
    int*   lists  = (int*)(ws + WS_LISTS);
    float* xemb   = (float*)(ws + WS_XEMB);
    float* xg     = (float*)(ws + WS_XGATES);

    setup_kernel<<<1, 256, 0, stream>>>(x, tags, emb, counts, lists, xemb);

    dim3 grid(G4 / 16, (T_LEN + 15) / 16, NEXP);
    xproj_kernel<<<grid, 32, 0, stream>>>(W_ih, b_ih, b_hh, counts, lists, xemb, xg);

    lstm_seq_kernel<<<1, 1024, 0, stream>>>(tags, h0, c0, W_hh, xg, fc_w, fc_b, out);
}